// GATLayer_13202729467970
// MI455X (gfx1250) — compile-verified
//
#include <hip/hip_runtime.h>
#include <math.h>

// ---------------------------------------------------------------------------
// 2-layer GAT for MI455X (gfx1250, wave32).
// GEMMs use V_WMMA_F32_16X16X4_F32 (fp32 matrix core path, full precision).
// Edge softmax + scatter use f32 global atomics (L2 atomic units).
// ---------------------------------------------------------------------------

typedef float v2f __attribute__((ext_vector_type(2)));
typedef float v8f __attribute__((ext_vector_type(8)));

#define GN0 120000
#define GN1 24000
#define GN2 6000
#define GD  256
#define GH1 8
#define GC1 64
#define GHC1 512
#define GH2 8
#define GC2 256
#define GHC2 2048
#define GE1 384000
#define GE2 96000
#define NEG_SLOPE 0.2f

// ---- float atomic max via sign-magnitude ordering trick (init to -inf) ----
__device__ __forceinline__ void atomicMaxF(float* addr, float v) {
  if (v >= 0.0f)
    atomicMax((int*)addr, __float_as_int(v));
  else
    atomicMin((unsigned int*)addr, __float_as_uint(v));
}

// ---------------------------------------------------------------------------
// WMMA fp32 GEMM: C[M,N] = A[M,K] * B[K,N].
// One wave per block; wave computes a 16(M) x 64(N) strip.
// Requires M%16==0, N%64==0, K%4==0 (true for both GEMMs here).
// A-matrix per-lane layout (16x4 f32): lanes 0-15 rows M=0..15 hold K={k,k+1},
// lanes 16-31 same rows hold K={k+2,k+3}. B symmetric (lanes = N columns).
// D (16x16 f32): VGPR r -> row r (lanes 0-15) / row r+8 (lanes 16-31).
// ---------------------------------------------------------------------------
__global__ void __launch_bounds__(32)
wmma_gemm_f32(const float* __restrict__ A, const float* __restrict__ B,
              float* __restrict__ C, int M, int N, int K) {
  const int lane  = threadIdx.x & 31;
  const int mrow  = lane & 15;   // row within tile (also N column for B/D)
  const int khalf = lane >> 4;   // 0: K pair {0,1}; 1: K pair {2,3}
  const int m0 = blockIdx.x << 4;  // M tiles on x (large dim)
  const int n0 = blockIdx.y << 6;  // N strip of 64 on y

  v8f acc[4];
  acc[0] = {}; acc[1] = {}; acc[2] = {}; acc[3] = {};

  const float* arow = A + (size_t)(m0 + mrow) * K + 2 * khalf;

  for (int k0 = 0; k0 < K; k0 += 4) {
    v2f a;
    a.x = arow[k0];
    a.y = arow[k0 + 1];
    const float* bp = B + (size_t)(k0 + 2 * khalf) * N + n0 + mrow;
#pragma unroll
    for (int j = 0; j < 4; ++j) {
      v2f b;
      b.x = bp[j * 16];
      b.y = bp[N + j * 16];
      acc[j] = __builtin_amdgcn_wmma_f32_16x16x4_f32(
          false, a, false, b, (short)0, acc[j], false, false);
    }
  }

#pragma unroll
  for (int j = 0; j < 4; ++j) {
    const int col = n0 + j * 16 + mrow;
#pragma unroll
    for (int r = 0; r < 8; ++r) {
      const int row = m0 + r + (khalf << 3);
      C[(size_t)row * N + col] = acc[j][r];
    }
  }
}

// ---------------------------------------------------------------------------
// a[n,h] = sum_c h[n,h,c] * att[h,c]. One wave per (node, head).
// ---------------------------------------------------------------------------
__global__ void att_reduce(const float* __restrict__ h,
                           const float* __restrict__ att,
                           float* __restrict__ out,
                           int n_nodes, int HH, int CC) {
  const int w = (int)((blockIdx.x * (size_t)blockDim.x + threadIdx.x) >> 5);
  const int lane = threadIdx.x & 31;
  if (w >= n_nodes * HH) return;           // whole wave exits together
  const int node = w / HH;
  const int head = w - node * HH;
  const float* hp = h + (size_t)node * HH * CC + (size_t)head * CC;
  const float* ap = att + (size_t)head * CC;
  float s = 0.0f;
  for (int c = lane; c < CC; c += 32) s += hp[c] * ap[c];
#pragma unroll
  for (int off = 16; off > 0; off >>= 1) s += __shfl_xor(s, off, 32);
  if (lane == 0) out[w] = s;
}

// ---------------------------------------------------------------------------
// Pass A: e = leakyrelu(a_src[src]+a_dst[dst]); store; atomic max per segment.
// ---------------------------------------------------------------------------
__global__ void edge_logits_max(const int* __restrict__ src,
                                const int* __restrict__ dst,
                                const float* __restrict__ a_src,
                                const float* __restrict__ a_dst,
                                float* __restrict__ ebuf,
                                float* __restrict__ m,
                                int E, int HH) {
  const int t = blockIdx.x * blockDim.x + threadIdx.x;
  if (t >= E * HH) return;
  const int e = t / HH;
  const int h = t - e * HH;
  float v = a_src[src[e] * HH + h] + a_dst[dst[e] * HH + h];
  v = (v > 0.0f) ? v : NEG_SLOPE * v;
  ebuf[t] = v;
  atomicMaxF(&m[dst[e] * HH + h], v);
}

// ---------------------------------------------------------------------------
// Pass B: ex = exp(e - m[dst]); store; atomic add into denom.
// ---------------------------------------------------------------------------
__global__ void edge_exp_sum(const int* __restrict__ dst,
                             float* __restrict__ ebuf,
                             const float* __restrict__ m,
                             float* __restrict__ denom,
                             int E, int HH) {
  const int t = blockIdx.x * blockDim.x + threadIdx.x;
  if (t >= E * HH) return;
  const int e = t / HH;
  const int h = t - e * HH;
  const float ex = expf(ebuf[t] - m[dst[e] * HH + h]);
  ebuf[t] = ex;
  atomicAdd(&denom[dst[e] * HH + h], ex);
}

// ---------------------------------------------------------------------------
// Pass C: out[dst, col] += scale * (ex/denom[dst,h]) * h[src, h*CC+c].
// One block per edge. concat: col = h*CC+c (outC == HH*CC); else col = c
// (head-mean folded in via scale = 1/HH). cshift = log2(CC).
// ---------------------------------------------------------------------------
__global__ void edge_scatter(const float* __restrict__ hsrc,
                             const int* __restrict__ src,
                             const int* __restrict__ dst,
                             const float* __restrict__ ex,
                             const float* __restrict__ denom,
                             float* __restrict__ out,
                             int HH, int cshift, int outC,
                             float scale, int concat) {
  const int e = blockIdx.x;
  const int s = src[e];
  const int d = dst[e];
  const int HCt = HH << cshift;
  const int cmask = (1 << cshift) - 1;
  for (int i = threadIdx.x; i < HCt; i += blockDim.x) {
    const int h = i >> cshift;
    const float alpha =
        scale * ex[(size_t)e * HH + h] / denom[(size_t)d * HH + h];
    const int col = concat ? i : (i & cmask);
    atomicAdd(&out[(size_t)d * outC + col],
              alpha * hsrc[(size_t)s * HCt + i]);
  }
}

// ---------------------------------------------------------------------------
// Small utility kernels.
// ---------------------------------------------------------------------------
__global__ void fill_f32(float* __restrict__ p, float v, long n) {
  const long i = blockIdx.x * (long)blockDim.x + threadIdx.x;
  if (i < n) p[i] = v;
}

__global__ void init_rows_bias(float* __restrict__ out,
                               const float* __restrict__ bias,
                               long rows, int cols) {
  const long t = blockIdx.x * (long)blockDim.x + threadIdx.x;
  if (t >= rows * cols) return;
  out[t] = bias[t % cols];
}

__global__ void elu_inplace(float* __restrict__ p, long n) {
  const long i = blockIdx.x * (long)blockDim.x + threadIdx.x;
  if (i >= n) return;
  const float v = p[i];
  p[i] = (v > 0.0f) ? v : expm1f(v);
}

// ---------------------------------------------------------------------------
// Launcher. Workspace needed: ~315 MB (h1 region reused for h2).
// ---------------------------------------------------------------------------
extern "C" void kernel_launch(void* const* d_in, const int* in_sizes, int n_in,
                              void* d_out, int out_size, void* d_ws,
                              size_t ws_size, hipStream_t stream) {
  (void)in_sizes; (void)n_in; (void)out_size; (void)ws_size;

  const float* x        = (const float*)d_in[0];
  const int*   src1     = (const int*)d_in[1];
  const int*   dst1     = (const int*)d_in[2];
  const int*   src2     = (const int*)d_in[3];
  const int*   dst2     = (const int*)d_in[4];
  const float* W1       = (const float*)d_in[5];
  const float* att_src1 = (const float*)d_in[6];
  const float* att_dst1 = (const float*)d_in[7];
  const float* b1       = (const float*)d_in[8];
  const float* W2       = (const float*)d_in[9];
  const float* att_src2 = (const float*)d_in[10];
  const float* att_dst2 = (const float*)d_in[11];
  const float* b2       = (const float*)d_in[12];
  float* out = (float*)d_out;

  char* ws = (char*)d_ws;
  size_t off = 0;
  auto alloc = [&](size_t bytes) -> float* {
    off = (off + 255) & ~(size_t)255;
    float* p = (float*)(ws + off);
    off += bytes;
    return p;
  };

  float* h1      = alloc((size_t)GN0 * GHC1 * 4);  // 245.8 MB; reused for h2
  float* x1      = alloc((size_t)GN1 * GHC1 * 4);  // 49.2 MB
  float* ebuf    = alloc((size_t)GE1 * GH1 * 4);   // 12.3 MB (>= E2*H2)
  float* a_src1  = alloc((size_t)GN0 * GH1 * 4);
  float* a_dst1  = alloc((size_t)GN1 * GH1 * 4);
  float* m1      = alloc((size_t)GN1 * GH1 * 4);
  float* denom1  = alloc((size_t)GN1 * GH1 * 4);
  float* a_src2  = alloc((size_t)GN1 * GH2 * 4);
  float* a_dst2  = alloc((size_t)GN2 * GH2 * 4);
  float* m2      = alloc((size_t)GN2 * GH2 * 4);
  float* denom2  = alloc((size_t)GN2 * GH2 * 4);
  float* h2      = h1;  // layer-1 intermediates dead by the time h2 is built

  // ---------------- Layer 1 ----------------
  // h1 = x @ W1  [120000 x 512], K = 256
  wmma_gemm_f32<<<dim3(GN0 / 16, GHC1 / 64), 32, 0, stream>>>(
      x, W1, h1, GN0, GHC1, GD);

  att_reduce<<<(GN0 * GH1 + 7) / 8, 256, 0, stream>>>(
      h1, att_src1, a_src1, GN0, GH1, GC1);
  att_reduce<<<(GN1 * GH1 + 7) / 8, 256, 0, stream>>>(
      h1, att_dst1, a_dst1, GN1, GH1, GC1);

  fill_f32<<<(GN1 * GH1 + 255) / 256, 256, 0, stream>>>(
      m1, -INFINITY, (long)GN1 * GH1);
  fill_f32<<<(GN1 * GH1 + 255) / 256, 256, 0, stream>>>(
      denom1, 0.0f, (long)GN1 * GH1);

  edge_logits_max<<<(GE1 * GH1 + 255) / 256, 256, 0, stream>>>(
      src1, dst1, a_src1, a_dst1, ebuf, m1, GE1, GH1);
  edge_exp_sum<<<(GE1 * GH1 + 255) / 256, 256, 0, stream>>>(
      dst1, ebuf, m1, denom1, GE1, GH1);

  init_rows_bias<<<((long)GN1 * GHC1 + 255) / 256, 256, 0, stream>>>(
      x1, b1, GN1, GHC1);
  edge_scatter<<<GE1, 128, 0, stream>>>(
      h1, src1, dst1, ebuf, denom1, x1, GH1, /*log2(C1)=*/6, GHC1,
      1.0f, /*concat=*/1);
  elu_inplace<<<((long)GN1 * GHC1 + 255) / 256, 256, 0, stream>>>(
      x1, (long)GN1 * GHC1);

  // ---------------- Layer 2 ----------------
  // h2 = x1 @ W2  [24000 x 2048], K = 512
  wmma_gemm_f32<<<dim3(GN1 / 16, GHC2 / 64), 32, 0, stream>>>(
      x1, W2, h2, GN1, GHC2, GHC1);

  att_reduce<<<(GN1 * GH2 + 7) / 8, 256, 0, stream>>>(
      h2, att_src2, a_src2, GN1, GH2, GC2);
  att_reduce<<<(GN2 * GH2 + 7) / 8, 256, 0, stream>>>(
      h2, att_dst2, a_dst2, GN2, GH2, GC2);

  fill_f32<<<(GN2 * GH2 + 255) / 256, 256, 0, stream>>>(
      m2, -INFINITY, (long)GN2 * GH2);
  fill_f32<<<(GN2 * GH2 + 255) / 256, 256, 0, stream>>>(
      denom2, 0.0f, (long)GN2 * GH2);

  edge_logits_max<<<(GE2 * GH2 + 255) / 256, 256, 0, stream>>>(
      src2, dst2, a_src2, a_dst2, ebuf, m2, GE2, GH2);
  edge_exp_sum<<<(GE2 * GH2 + 255) / 256, 256, 0, stream>>>(
      dst2, ebuf, m2, denom2, GE2, GH2);

  // out initialized to bias rows; head-mean folded into scale = 1/H2.
  init_rows_bias<<<((long)GN2 * GC2 + 255) / 256, 256, 0, stream>>>(
      out, b2, GN2, GC2);
  edge_scatter<<<GE2, 256, 0, stream>>>(
      h2, src2, dst2, ebuf, denom2, out, GH2, /*log2(C2)=*/8, GC2,
      1.0f / (float)GH2, /*concat=*/0);
}